// ForwardKinematics_38543036514646
// MI455X (gfx1250) — compile-verified
//
#include <hip/hip_runtime.h>

typedef float v2f __attribute__((ext_vector_type(2)));
typedef float v8f __attribute__((ext_vector_type(8)));

#define BB 64
#define TT 1024
#define JJ 52
#define CPB 4              // chains (b,t pairs) per block == per wave
#define CSTRIDE (JJ * 16)  // 832 floats per chain (52 x 4x4)

__global__ __launch_bounds__(32) void fk_wmma_kernel(
    const float* __restrict__ rotation,   // (B,T,J,3,3)
    const float* __restrict__ position,   // (B,T,3)
    const float* __restrict__ offsets,    // (B,J,3)
    const int*   __restrict__ parents,    // (1,J)
    float* __restrict__ out_pos,          // (B,T,J,3)
    float* __restrict__ out_comp)         // (B,T,J,4,4)
{
  __shared__ float loc [CPB * CSTRIDE];       // local 4x4 transforms, row-major
  __shared__ float comp[CPB * CSTRIDE];       // composed 4x4 transforms
  __shared__ __align__(16) float dtile[256];  // one 16x16 WMMA D tile, col-major per lane

  const int  lane   = threadIdx.x;                    // 0..31
  const long chain0 = (long)blockIdx.x * CPB;         // first (b,t) id of block

  // ---- Stage local transforms into LDS (mostly-coalesced monotone reads) ----
  for (int idx = lane; idx < CPB * CSTRIDE; idx += 32) {
    int chain = idx / CSTRIDE;
    int rem   = idx - chain * CSTRIDE;
    int j = rem >> 4;
    int e = rem & 15;
    int r = e >> 2;
    int c = e & 3;
    long id = chain0 + chain;
    int  b  = (int)(id >> 10);          // T = 1024
    int  t  = (int)(id & 1023);
    float v;
    if (r == 3) {
      v = (c == 3) ? 1.0f : 0.0f;
    } else if (c == 3) {
      v = (j == 0) ? position[((long)b * TT + t) * 3 + r]
                   : offsets [((long)b * JJ + j) * 3 + r];
    } else {
      v = rotation[(((long)b * TT + t) * JJ + j) * 9 + r * 3 + c];
    }
    loc[idx] = v;
  }
  __syncthreads();

  // ---- joint 0: composed = local ----
  for (int idx = lane; idx < CPB * 16; idx += 32) {
    int chain = idx >> 4;
    int e     = idx & 15;
    comp[chain * CSTRIDE + e] = loc[chain * CSTRIDE + e];
  }

  // Per-lane constants for WMMA operand layouts (f32 16x16x4).
  // A (16x4): lane l vgpr0 = A[l%16][l<16?0:2], vgpr1 = A[l%16][l<16?1:3]
  // B (4x16): lane l vgpr0 = B[l<16?0:2][l%16], vgpr1 = B[l<16?1:3][l%16]
  // D (16x16): lane n(0-15) vgpr v = D[v][n]; lane 16+n vgpr v = D[v+8][n]
  const int  m  = lane & 15;
  const int  k0 = (lane < 16) ? 0 : 2;
  const int  q  = m >> 2;          // diagonal 4x4 block index (chain)
  const int  c4 = m & 3;           // row (A) / col (B) within block

  // Loop-invariant A/B LDS offsets (add p*16 / j*16 per iteration).
  const int aOff = q * CSTRIDE + c4 * 4 + k0;        // + p*16
  const int bOff = q * CSTRIDE + k0 * 4 + c4;        // + j*16   (pair at +16B)

  // Extraction plan: 64 diagonal elements, 2 per lane, addresses loop-invariant.
  // idx -> (eq, err, ec4); row = 4eq+err, col = 4eq+ec4
  // dtile src = (col + 16*(row>>3))*8 + (row&7)
  // comp dst  = eq*CSTRIDE + err*4 + ec4   (+ j*16)
  int s0, d0, s1, d1;
  {
    int idx = lane;
    int eq = idx >> 4, err = (idx >> 2) & 3, ec4 = idx & 3;
    int row = 4 * eq + err, col = 4 * eq + ec4;
    s0 = (col + ((row >> 3) << 4)) * 8 + (row & 7);
    d0 = eq * CSTRIDE + err * 4 + ec4;
    idx = lane + 32;
    eq = idx >> 4; err = (idx >> 2) & 3; ec4 = idx & 3;
    row = 4 * eq + err; col = 4 * eq + ec4;
    s1 = (col + ((row >> 3) << 4)) * 8 + (row & 7);
    d1 = eq * CSTRIDE + err * 4 + ec4;
  }
  __syncthreads();

  for (int j = 1; j < JJ; ++j) {
    const int p = parents[j];   // uniform -> scalar load (constant cache)

    // A = vertical stack of 4 parent matrices: A[4q+r][k] = comp[q][p][r][k]
    float a0 = comp[aOff + p * 16];
    float a1 = comp[aOff + p * 16 + 1];
    // B = horizontal stack of 4 local matrices: B[k][4q+c] = loc[q][j][k][c]
    float b0 = loc [bOff + j * 16];
    float b1 = loc [bOff + j * 16 + 4];

    v2f A; A.x = a0; A.y = a1;
    v2f Bv; Bv.x = b0; Bv.y = b1;
    v8f C = {};
    v8f D = __builtin_amdgcn_wmma_f32_16x16x4_f32(
        /*neg_a=*/false, A, /*neg_b=*/false, Bv,
        /*c_mod=*/(short)0, C, /*reuse_a=*/false, /*reuse_b=*/false);

    // Dump the whole D tile to LDS (2x ds_store_b128 per lane), no lane branches.
    float4* dst = reinterpret_cast<float4*>(&dtile[lane * 8]);
    float4 lo, hi;
    lo.x = D[0]; lo.y = D[1]; lo.z = D[2]; lo.w = D[3];
    hi.x = D[4]; hi.y = D[5]; hi.z = D[6]; hi.w = D[7];
    dst[0] = lo;
    dst[1] = hi;
    __syncthreads();

    // Gather the 4x4 diagonal blocks into comp (2 elements per lane).
    comp[d0 + j * 16] = dtile[s0];
    comp[d1 + j * 16] = dtile[s1];
    __syncthreads();
  }

  // ---- Writeout: LDS layout == global layout -> coalesced b128 stream ----
  {
    const float4* csrc = reinterpret_cast<const float4*>(comp);
    float4* cdst = reinterpret_cast<float4*>(out_comp + chain0 * (JJ * 16));
    for (int idx = lane; idx < CPB * CSTRIDE / 4; idx += 32)
      cdst[idx] = csrc[idx];
  }

  const long pos_base = chain0 * (JJ * 3);
  for (int idx = lane; idx < CPB * JJ * 3; idx += 32) {
    int chain = idx / (JJ * 3);
    int rem   = idx - chain * (JJ * 3);
    int j = rem / 3;
    int r = rem - j * 3;
    out_pos[pos_base + idx] = comp[chain * CSTRIDE + j * 16 + r * 4 + 3];
  }
}

extern "C" void kernel_launch(void* const* d_in, const int* in_sizes, int n_in,
                              void* d_out, int out_size, void* d_ws, size_t ws_size,
                              hipStream_t stream) {
  (void)in_sizes; (void)n_in; (void)out_size; (void)d_ws; (void)ws_size;
  const float* rotation = (const float*)d_in[0];
  const float* position = (const float*)d_in[1];
  const float* offsets  = (const float*)d_in[2];
  const int*   parents  = (const int*)d_in[3];

  float* out_pos  = (float*)d_out;                          // B*T*J*3 floats
  float* out_comp = out_pos + (long)BB * TT * JJ * 3;       // then B*T*J*16

  dim3 grid(BB * TT / CPB);   // 16384 blocks, 1 wave each (4 chains/wave)
  fk_wmma_kernel<<<grid, 32, 0, stream>>>(rotation, position, offsets, parents,
                                          out_pos, out_comp);
}